// Node_1760936591969
// MI455X (gfx1250) — compile-verified
//
#include <hip/hip_runtime.h>

typedef int   v4i __attribute__((ext_vector_type(4)));
typedef float v4f __attribute__((ext_vector_type(4)));

#define THREADS 256

// Hardware float atomic-add, no return (STOREcnt path, resolved in L2).
__device__ __forceinline__ void atomic_add_f32_noret(float* p, float v) {
    asm volatile("global_atomic_add_f32 %0, %1, off"
                 :
                 : "v"(p), "v"(v)
                 : "memory");
}

// CDNA5 hardware tanh (V_TANH_F32, single TRANS32 op). Guarded fallback keeps
// the build safe if the builtin name differs on this toolchain.
__device__ __forceinline__ float hw_tanh(float x) {
#if __has_builtin(__builtin_amdgcn_tanhf)
    return __builtin_amdgcn_tanhf(x);
#else
    // tanh(x) = 1 - 2/(e^{2x}+1); exact limits at +/-inf, monotone, cheap.
    float e = __expf(2.0f * x);
    return 1.0f - 2.0f / (e + 1.0f);
#endif
}

// V_SIN_F32 path: __sinf lowers to the hardware sin after 1/(2*pi) scaling.
__device__ __forceinline__ float hw_sin(float x) {
    return __sinf(x);
}

__global__ void zero_kernel(float* __restrict__ agg, int n) {
    int i = blockIdx.x * blockDim.x + threadIdx.x;
    if (i < n) agg[i] = 0.0f;
}

// msg = node_output[edge_src] * edge_weight; agg[edge_dst] += msg
// 4 edges per thread via 128-bit non-temporal loads (edge streams are
// touch-once: NT keeps them from evicting the L2-resident node_output/agg).
__global__ void edge_kernel(const float* __restrict__ node_output,
                            const float* __restrict__ edge_weight,
                            const int*   __restrict__ edge_src,
                            const int*   __restrict__ edge_dst,
                            float* __restrict__ agg,
                            int n_edges) {
    int n4  = n_edges >> 2;
    int tid = blockIdx.x * blockDim.x + threadIdx.x;

    if (tid < n4) {
        v4i src = __builtin_nontemporal_load((const v4i*)edge_src + tid);
        v4f w   = __builtin_nontemporal_load((const v4f*)edge_weight + tid);
        v4i dst = __builtin_nontemporal_load((const v4i*)edge_dst + tid);

        float m0 = node_output[src.x] * w.x;
        float m1 = node_output[src.y] * w.y;
        float m2 = node_output[src.z] * w.z;
        float m3 = node_output[src.w] * w.w;

        atomic_add_f32_noret(agg + dst.x, m0);
        atomic_add_f32_noret(agg + dst.y, m1);
        atomic_add_f32_noret(agg + dst.z, m2);
        atomic_add_f32_noret(agg + dst.w, m3);
    }

    // Tail (n_edges not a multiple of 4): first (n_edges & 3) threads handle it.
    int tail_base = n4 << 2;
    int rem = n_edges - tail_base;
    if (tid < rem) {
        int e = tail_base + tid;
        float m = node_output[edge_src[e]] * edge_weight[e];
        atomic_add_f32_noret(agg + edge_dst[e], m);
    }
}

// x = agg + p0 ; out = p1*tanh(x)*sin(p2*x + p3) + p4*x + p5
__global__ void node_kernel(const float* __restrict__ agg,
                            const float* __restrict__ node_params,
                            float* __restrict__ out,
                            int n_nodes) {
    int i = blockIdx.x * blockDim.x + threadIdx.x;
    if (i >= n_nodes) return;

    const float* p = node_params + (size_t)i * 7;
    float p0 = p[0];
    float p1 = p[1];
    float p2 = p[2];
    float p3 = p[3];
    float p4 = p[4];
    float p5 = p[5];

    float x = agg[i] + p0;
    float t = hw_tanh(x);
    float s = hw_sin(fmaf(p2, x, p3));
    float r = fmaf(p1 * t, s, fmaf(p4, x, p5));

    __builtin_nontemporal_store(r, out + i);
}

extern "C" void kernel_launch(void* const* d_in, const int* in_sizes, int n_in,
                              void* d_out, int out_size, void* d_ws, size_t ws_size,
                              hipStream_t stream) {
    const float* node_output = (const float*)d_in[0];
    const float* edge_weight = (const float*)d_in[1];
    const float* node_params = (const float*)d_in[2];
    const int*   edge_src    = (const int*)d_in[3];
    const int*   edge_dst    = (const int*)d_in[4];
    float*       out         = (float*)d_out;

    int n_nodes = in_sizes[0];
    int n_edges = in_sizes[1];

    float* agg = (float*)d_ws;  // n_nodes * 4 bytes of scratch (4 MB)

    int node_blocks = (n_nodes + THREADS - 1) / THREADS;
    int n4 = n_edges >> 2;
    int rem = n_edges & 3;
    int edge_work = (n4 > rem) ? n4 : rem;
    int edge_blocks = (edge_work + THREADS - 1) / THREADS;
    if (edge_blocks < 1) edge_blocks = 1;

    zero_kernel<<<node_blocks, THREADS, 0, stream>>>(agg, n_nodes);
    edge_kernel<<<edge_blocks, THREADS, 0, stream>>>(node_output, edge_weight,
                                                     edge_src, edge_dst,
                                                     agg, n_edges);
    node_kernel<<<node_blocks, THREADS, 0, stream>>>(agg, node_params, out, n_nodes);
}